// Warp3D_44890998178347
// MI455X (gfx1250) — compile-verified
//
#include <hip/hip_runtime.h>

#ifndef __has_builtin
#define __has_builtin(x) 0
#endif

// Problem constants (from reference: B=4, C_in=16, T=16, H=256, W=256)
#define WP_T   16
#define WP_H   256
#define WP_W   256
#define WP_C   16
#define WP_HW  (WP_H * WP_W)        // 65536
#define WP_THW (WP_T * WP_HW)       // 1048576

#if __has_builtin(__builtin_amdgcn_global_load_async_to_lds_b32)
#define USE_ASYNC_LDS 1
#else
#define USE_ASYNC_LDS 0
#endif

__global__ __launch_bounds__(256) void warp_bilinear_kernel(
    const float* __restrict__ input,   // [B, C, T, H, W]
    const float* __restrict__ flows,   // [B, 2, T, H, W]
    float* __restrict__ out,           // [B, C, T, H, W]
    int npix)                          // B*T*H*W
{
    const int tid = threadIdx.x;
    const int gid = blockIdx.x * 256 + tid;
    if (gid >= npix) return;

    // gid = ((b*T + t)*H + h)*W + w   (H, W are powers of two)
    const int w  = gid & (WP_W - 1);
    const int h  = (gid >> 8) & (WP_H - 1);
    const int bt = gid >> 16;           // b*T + t
    const int b  = bt >> 4;
    const int t  = bt & (WP_T - 1);

    // flows index of channel 0 at (b,t,h,w); channel 1 is +THW
    const int fidx = (b * 2) * WP_THW + t * WP_HW + h * WP_W + w;

    float f0, f1;
#if USE_ASYNC_LDS
    // CDNA5 async global->LDS (ASYNCcnt) staging of the flow pair.
    // Each lane writes/reads only its own LDS slot, so s_wait_asynccnt alone
    // (no barrier) is sufficient.
    __shared__ float sflow[2][256];
    typedef __attribute__((address_space(1))) int gint_t;  // clang-reported param type
    typedef __attribute__((address_space(3))) int lint_t;
    __builtin_amdgcn_global_load_async_to_lds_b32(
        (gint_t*)(float*)(flows + fidx),
        (lint_t*)(void*)&sflow[0][tid], 0, 0);
    __builtin_amdgcn_global_load_async_to_lds_b32(
        (gint_t*)(float*)(flows + fidx + WP_THW),
        (lint_t*)(void*)&sflow[1][tid], 0, 0);
#if __has_builtin(__builtin_amdgcn_s_wait_asynccnt)
    __builtin_amdgcn_s_wait_asynccnt(0);
#else
    asm volatile("s_wait_asynccnt 0" ::: "memory");
#endif
    f0 = sflow[0][tid];
    f1 = sflow[1][tid];
#else
    // Flows are read exactly once: non-temporal so they don't pollute L2,
    // which we want to keep for the gathered input frames.
    f0 = __builtin_nontemporal_load(flows + fidx);
    f1 = __builtin_nontemporal_load(flows + fidx + WP_THW);
#endif

    // Reference math collapses exactly:
    //   fx = flow0 / H * 2 ;  ix = clip(w + fx*0.5*(W-1), 0, W-1)
    //   => ix = clip(w + flow0 * (2/256)*127.5, 0, 255), same for iy.
    const float SC = 0.99609375f;  // (2/256) * 0.5 * 255
    float ix = fminf(fmaxf(fmaf(f0, SC, (float)w), 0.0f), (float)(WP_W - 1));
    float iy = fminf(fmaxf(fmaf(f1, SC, (float)h), 0.0f), (float)(WP_H - 1));

    const float x0f = floorf(ix);
    const float y0f = floorf(iy);
    const float wx = ix - x0f;
    const float wy = iy - y0f;
    const int x0 = (int)x0f;
    const int y0 = (int)y0f;
    const int x1 = min(x0 + 1, WP_W - 1);
    const int y1 = min(y0 + 1, WP_H - 1);

    const float omwx = 1.0f - wx;
    const float omwy = 1.0f - wy;
    const float w00 = omwy * omwx;
    const float w01 = omwy * wx;
    const float w10 = wy * omwx;
    const float w11 = wy * wx;

    const int i00 = y0 * WP_W + x0;
    const int i01 = y0 * WP_W + x1;
    const int i10 = y1 * WP_W + x0;
    const int i11 = y1 * WP_W + x1;

    // Per-(b,t) frame base for channel 0; channel stride is THW in both
    // input and output.
    const float* __restrict__ pin =
        input + (size_t)(b * WP_C) * WP_THW + (size_t)t * WP_HW;
    float* __restrict__ pout =
        out + (size_t)(b * WP_C) * WP_THW + (size_t)t * WP_HW
            + h * WP_W + w;

    // One set of weights/indices amortized over all 16 channels.
#pragma unroll
    for (int c = 0; c < WP_C; ++c) {
        const float v00 = pin[i00];   // cached gathers (want temporal)
        const float v01 = pin[i01];
        const float v10 = pin[i10];
        const float v11 = pin[i11];
        const float r = fmaf(v11, w11, fmaf(v10, w10, fmaf(v01, w01, v00 * w00)));
        // Output is write-once streaming: NT store keeps L2 for the input.
        __builtin_nontemporal_store(r, pout);
        pin  += WP_THW;
        pout += WP_THW;
    }
}

extern "C" void kernel_launch(void* const* d_in, const int* in_sizes, int n_in,
                              void* d_out, int out_size, void* d_ws, size_t ws_size,
                              hipStream_t stream) {
    const float* input = (const float*)d_in[0];  // [B, C, T, H, W]
    const float* flows = (const float*)d_in[1];  // [B, 2, T, H, W]
    float* out = (float*)d_out;                  // [B, C, T, H, W]

    const int npix = out_size / WP_C;            // B*T*H*W = 4,194,304
    const int blocks = (npix + 255) / 256;       // 16384 blocks x 8 wave32
    warp_bilinear_kernel<<<blocks, 256, 0, stream>>>(input, flows, out, npix);
}